// MarkedPointProcess_72902774882536
// MI455X (gfx1250) — compile-verified
//
#include <hip/hip_runtime.h>
#include <hip/hip_bf16.h>
#include <stdint.h>

// Hawkes-process intensity: out[j] = softplus(base[j] + sum_i alpha[k_i,j]*exp(-beta[k_i,j]*dt_i))
// N ~ 1e6 events, K = 128. Compute-bound on v_exp_f32 (N*K ~ 1.28e8 exps); memory is negligible
// (8 MB events at 23.3 TB/s). Strategy: alpha/beta resident in LDS, events streamed via CDNA5
// async global->LDS loads (double buffered), lane == output column. Final cross-block reduction
// done with V_WMMA_F32_16X16X4_F32 (ones-matrix column sum) for a deterministic f32 reduce.

#define KTYPES 128
#define NB     128   // accumulation blocks (also rows of the partial matrix; multiple of 4)
#define TILE   256   // events staged per async tile (== blockDim)
#define BLOCK  256

typedef __attribute__((ext_vector_type(2))) float v2f;
typedef __attribute__((ext_vector_type(8))) float v8f;

__device__ __forceinline__ unsigned lds_off(const void* p) {
  // Low 32 bits of a flat pointer into the LDS aperture are the LDS byte offset.
  return (unsigned)(uintptr_t)p;
}

__device__ __forceinline__ void async_load_b32(unsigned ldsdst, const void* gsrc) {
  asm volatile("global_load_async_to_lds_b32 %0, %1, off"
               :: "v"(ldsdst), "v"(gsrc) : "memory");
}

__global__ __launch_bounds__(BLOCK) void hawkes_accum(
    const float* __restrict__ times, const int* __restrict__ types,
    const float* __restrict__ alpha, const float* __restrict__ beta,
    float* __restrict__ partials, int n, int per_block)
{
  __shared__ float s_alpha[KTYPES * KTYPES];   // 64 KB
  __shared__ float s_beta [KTYPES * KTYPES];   // 64 KB
  __shared__ float s_time[2][TILE];            // double-buffered event times
  __shared__ int   s_type[2][TILE];            // double-buffered event types
  __shared__ float s_red[KTYPES];

  const int tid   = threadIdx.x;
  const int m     = n - 1;                       // events participating ([:-1])
  const int start = blockIdx.x * per_block;
  const int end   = min(m, start + per_block);
  const int len   = max(0, end - start);
  const int tiles = (len + TILE - 1) / TILE;

  // Kick off async tile 0 immediately; it overlaps with alpha/beta staging below.
  if (tiles > 0) {
    const int idx = min(start + tid, m - 1);     // clamp to stay in-bounds (extras never read)
    async_load_b32(lds_off(&s_time[0][tid]), &times[idx]);
    async_load_b32(lds_off(&s_type[0][tid]), &types[idx]);
  }

  // Stage alpha & beta into LDS (vectorized float4 copies).
  {
    const float4* a4 = (const float4*)alpha;
    const float4* b4 = (const float4*)beta;
    float4* sa4 = (float4*)s_alpha;
    float4* sb4 = (float4*)s_beta;
    for (int i = tid; i < (KTYPES * KTYPES) / 4; i += BLOCK) {
      sa4[i] = a4[i];
      sb4[i] = b4[i];
    }
  }

  const float Tlast = times[n - 1];
  const int   j   = tid & (KTYPES - 1);          // output column owned by this lane
  const int   par = tid >> 7;                    // 0/1: even/odd events within a tile
  float acc = 0.0f;

  for (int t = 0; t < tiles; ++t) {
    const int buf = t & 1;
    if (t + 1 < tiles) {
      // Prefetch next tile into the other buffer, then wait for the current tile
      // (async loads retire in order; <=2 outstanding leaves only the new pair in flight).
      const int idx = min(start + (t + 1) * TILE + tid, m - 1);
      async_load_b32(lds_off(&s_time[buf ^ 1][tid]), &times[idx]);
      async_load_b32(lds_off(&s_type[buf ^ 1][tid]), &types[idx]);
      asm volatile("s_wait_asynccnt 2" ::: "memory");
    } else {
      asm volatile("s_wait_asynccnt 0" ::: "memory");
    }
    __syncthreads();                             // all waves' tile-t data (and alpha/beta) visible

    const int tlen = min(TILE, len - t * TILE);
    for (int e = par; e < tlen; e += 2) {        // e is wave-uniform -> broadcast LDS reads
      const float dt = Tlast - s_time[buf][e];
      const int   k  = s_type[buf][e];
      const float a  = s_alpha[(k << 7) + j];    // lanes read consecutive j: conflict-free
      const float b  = s_beta [(k << 7) + j];
      if (dt > 0.0f) acc = fmaf(a, __expf(-b * dt), acc);
    }
    __syncthreads();                             // buffer reusable before next prefetch issue
  }

  // Combine the two event-parity partials and emit this block's row.
  if (par == 1) s_red[j] = acc;
  __syncthreads();
  if (par == 0) partials[blockIdx.x * KTYPES + j] = acc + s_red[j];
}

// Column-sum the NB x 128 partial matrix with V_WMMA_F32_16X16X4_F32:
// A = ones(16x4) so D[m][n] = sum_k B[k][n] + C[m][n]. Each of 8 waves owns 16 columns.
__global__ __launch_bounds__(256) void hawkes_reduce(
    const float* __restrict__ partials, const float* __restrict__ base,
    float* __restrict__ out)
{
  const int tid  = threadIdx.x;
  const int wave = tid >> 5;
  const int lane = tid & 31;
  const int col  = (wave << 4) + (lane & 15);
  const int rsel = (lane >> 4) << 1;             // lanes 0-15: rows b,b+1; lanes 16-31: b+2,b+3

  v2f a; a[0] = 1.0f; a[1] = 1.0f;               // all-ones A matrix (16x4)
  v8f c = {};
  for (int b = 0; b < NB; b += 4) {
    v2f bm;
    bm[0] = partials[(b + rsel + 0) * KTYPES + col];
    bm[1] = partials[(b + rsel + 1) * KTYPES + col];
    c = __builtin_amdgcn_wmma_f32_16x16x4_f32(
        /*neg_a=*/false, a, /*neg_b=*/false, bm,
        /*c_mod=*/(short)0, c, /*reuse_a=*/false, /*reuse_b=*/false);
  }
  if (lane < 16) {                               // C/D layout: VGPR0, lanes 0-15 -> M=0, N=lane
    const float x = base[col] + c[0];
    out[col] = fmaxf(x, 0.0f) + log1pf(expf(-fabsf(x)));   // stable softplus
  }
}

extern "C" void kernel_launch(void* const* d_in, const int* in_sizes, int n_in,
                              void* d_out, int out_size, void* d_ws, size_t ws_size,
                              hipStream_t stream) {
  (void)n_in; (void)out_size; (void)ws_size;
  // Input order: h(unused), event_times, event_types, base_intensity, alpha, beta
  const float* times = (const float*)d_in[1];
  const int*   types = (const int*)d_in[2];
  const float* base  = (const float*)d_in[3];
  const float* alpha = (const float*)d_in[4];
  const float* beta  = (const float*)d_in[5];
  float* out      = (float*)d_out;
  float* partials = (float*)d_ws;                // NB * 128 floats = 64 KB scratch

  const int n = in_sizes[1];
  const int m = n - 1;
  const int per_block = (m + NB - 1) / NB;

  hawkes_accum<<<NB, BLOCK, 0, stream>>>(times, types, alpha, beta, partials, n, per_block);
  hawkes_reduce<<<1, 256, 0, stream>>>(partials, base, out);
}